// QLSTM_35880156791526
// MI455X (gfx1250) — compile-verified
//
#include <hip/hip_runtime.h>

// QLSTM: B=256, T=512, F=128, H=256, NQ=4
#define QB 256
#define QT 512
#define QF 128
#define QH 256

typedef __attribute__((ext_vector_type(2))) float v2f;
typedef __attribute__((ext_vector_type(8))) float v8f;

// ---------------------------------------------------------------------------
// Kernel 1: Xproj[b,t,0:4] = x[b,t,:] @ W_in[H:H+F, 0:4]
// Bandwidth-bound GEMM [131072 x 128] @ [128 x 4], done with f32 WMMA 16x16x4.
// One wave handles one 16-row tile; N padded to 16 (cols 4..15 discarded).
// ---------------------------------------------------------------------------
__global__ __launch_bounds__(256) void qlstm_xproj_wmma(
    const float* __restrict__ X,    // [B*T, F] row-major
    const float* __restrict__ Wx,   // [F, 4]  (= W_in + H*4)
    float* __restrict__ Xproj)      // [B*T, 4]
{
    const int lane = threadIdx.x & 31;
    const int wave = threadIdx.x >> 5;
    const int tile = blockIdx.x * 8 + wave;       // 16-row tile index
    const int base = tile * 16;

    const int m  = lane & 15;                     // row within tile (A operand)
    const int kk = (lane >> 4) << 1;              // K sub-offset: 0 or 2
    const float* xrow = X + (size_t)(base + m) * QF + kk;

    // B operand: lane covers column n = lane&15 (valid only n<4), rows kk,kk+1
    const int   n   = lane & 15;
    const float msk = (n < 4) ? 1.0f : 0.0f;
    const float* wp = Wx + kk * 4 + (lane & 3);   // Wx[(kc*4+kk)*4 + n] = wp[kc*16]

    v8f acc = {};
#pragma unroll
    for (int kc = 0; kc < 32; ++kc) {             // K = 128 in chunks of 4
        v2f a = *(const v2f*)(xrow + kc * 4);     // A[m][kk], A[m][kk+1]
        v2f b;
        b.x = wp[kc * 16]     * msk;              // B[kk  ][n]
        b.y = wp[kc * 16 + 4] * msk;              // B[kk+1][n]
        // D = A(16x4) * B(4x16) + C  -> v_wmma_f32_16x16x4_f32
        acc = __builtin_amdgcn_wmma_f32_16x16x4_f32(
            /*neg_a=*/false, a, /*neg_b=*/false, b,
            /*c_mod=*/(short)0, acc, /*reuse_a=*/false, /*reuse_b=*/false);
    }

    // C layout: VGPR g -> lanes0-15: M=g, N=lane ; lanes16-31: M=8+g, N=lane-16
    const int col  = lane & 15;
    const int madd = (lane >> 4) * 8;
    if (col < 4) {
#pragma unroll
        for (int g = 0; g < 8; ++g)
            Xproj[(size_t)(base + g + madd) * 4 + col] = acc[g];
    }
}

// ---------------------------------------------------------------------------
// Fast transcendentals on v_exp_f32 (exp2) / v_rcp_f32
// ---------------------------------------------------------------------------
__device__ __forceinline__ float fast_sigmoid(float x) {
    return __builtin_amdgcn_rcpf(1.0f + __builtin_amdgcn_exp2f(-1.4426950408889634f * x));
}
__device__ __forceinline__ float fast_tanh(float x) {
    // tanh(x) = 2*sigmoid(2x) - 1 ; saturates correctly at +-1 for large |x|
    return __builtin_fmaf(2.0f,
        __builtin_amdgcn_rcpf(1.0f + __builtin_amdgcn_exp2f(-2.8853900817779268f * x)),
        -1.0f);
}

// ---------------------------------------------------------------------------
// Kernel 2: sequential recurrence. 1 workgroup = 1 batch element.
// thread tid owns hidden unit tid. One barrier per timestep (double-buffered
// LDS reduction slots). h/c live in registers.
// ---------------------------------------------------------------------------
__global__ __launch_bounds__(256) void qlstm_recur(
    const float* __restrict__ Xproj,   // [B, T, 4]
    const float* __restrict__ W_in,    // [H+F, 4]; rows 0..H-1 are the h-part
    const float* __restrict__ b_in,    // [4]
    const float* __restrict__ Wq,      // [4, 4, 4]  (k, q, r)
    const float* __restrict__ bq,      // [4, 4]
    const float* __restrict__ W_out,   // [4, H]
    const float* __restrict__ b_out,   // [H]
    float* __restrict__ hidden_seq,    // [B, T, H]
    float* __restrict__ hT,            // [B, H]
    float* __restrict__ cT)            // [B, H]
{
    const int tid  = threadIdx.x;      // hidden index 0..255
    const int b    = blockIdx.x;
    const int lane = tid & 31;
    const int wave = tid >> 5;

    __shared__ float4 red[2][8];       // double-buffered per-wave partials

    // Per-thread constants
    const float4 winh = *(const float4*)(W_in + tid * 4);        // W_in[tid, 0:4]
    const float  wo0 = W_out[tid], wo1 = W_out[QH + tid];
    const float  wo2 = W_out[2 * QH + tid], wo3 = W_out[3 * QH + tid];
    const float  bo = b_out[tid];

    // Uniform small tensors (scalar-loaded by compiler)
    float wq[64], bqr[16], bin[4];
#pragma unroll
    for (int i = 0; i < 64; ++i) wq[i] = Wq[i];
#pragma unroll
    for (int i = 0; i < 16; ++i) bqr[i] = bq[i];
#pragma unroll
    for (int i = 0; i < 4; ++i)  bin[i] = b_in[i];

    const float* xp = Xproj + (size_t)b * QT * 4;
    float* hs = hidden_seq + (size_t)b * QT * QH + tid;

    float h = 0.0f, c = 0.0f;

    for (int t = 0; t < QT; ++t) {
        // ---- y = h @ W_in_h : per-thread partials, wave32 butterfly reduce
        float p0 = h * winh.x, p1 = h * winh.y, p2 = h * winh.z, p3 = h * winh.w;
#pragma unroll
        for (int off = 16; off > 0; off >>= 1) {
            p0 += __shfl_xor(p0, off);
            p1 += __shfl_xor(p1, off);
            p2 += __shfl_xor(p2, off);
            p3 += __shfl_xor(p3, off);
        }
        if (lane == 0) red[t & 1][wave] = make_float4(p0, p1, p2, p3);

        // prefetch the Xproj stream a bit ahead (global_prefetch_b8)
        __builtin_prefetch(xp + (t + 16) * 4, 0, 1);

        __syncthreads();

        const float4 xv = *(const float4*)(xp + t * 4);   // uniform load
        float y0 = bin[0] + xv.x, y1 = bin[1] + xv.y;
        float y2 = bin[2] + xv.z, y3 = bin[3] + xv.w;
#pragma unroll
        for (int w = 0; w < 8; ++w) {
            float4 r = red[t & 1][w];
            y0 += r.x; y1 += r.y; y2 += r.z; y3 += r.w;
        }

        // ---- q[k][r] = tanh(bq + y@Wq[k]);  z[k] = q[k]@W_out[:,tid] + b_out
        float zg[4];
#pragma unroll
        for (int k = 0; k < 4; ++k) {
            float zk = bo;
#pragma unroll
            for (int r = 0; r < 4; ++r) {
                float a = bqr[k * 4 + r];
                a = __builtin_fmaf(y0, wq[k * 16 +  0 + r], a);
                a = __builtin_fmaf(y1, wq[k * 16 +  4 + r], a);
                a = __builtin_fmaf(y2, wq[k * 16 +  8 + r], a);
                a = __builtin_fmaf(y3, wq[k * 16 + 12 + r], a);
                float qv = fast_tanh(a);
                const float wo = (r == 0) ? wo0 : (r == 1) ? wo1 : (r == 2) ? wo2 : wo3;
                zk = __builtin_fmaf(qv, wo, zk);
            }
            zg[k] = zk;
        }

        // ---- LSTM cell update
        const float ig = fast_sigmoid(zg[0]);
        const float fg = fast_sigmoid(zg[1]);
        const float gg = fast_tanh(zg[2]);
        const float og = fast_sigmoid(zg[3]);
        c = __builtin_fmaf(fg, c, ig * gg);
        h = og * fast_tanh(c);

        hs[(size_t)t * QH] = h;   // coalesced, fire-and-forget
    }

    hT[(size_t)b * QH + tid] = h;
    cT[(size_t)b * QH + tid] = c;
}

// ---------------------------------------------------------------------------
extern "C" void kernel_launch(void* const* d_in, const int* in_sizes, int n_in,
                              void* d_out, int out_size, void* d_ws, size_t ws_size,
                              hipStream_t stream) {
    const float* x     = (const float*)d_in[0];  // [B,T,F]
    const float* W_in  = (const float*)d_in[1];  // [H+F, 4]
    const float* b_in  = (const float*)d_in[2];  // [4]
    const float* Wq    = (const float*)d_in[3];  // [4,4,4]
    const float* bq    = (const float*)d_in[4];  // [4,4]
    const float* W_out = (const float*)d_in[5];  // [4,H]
    const float* b_out = (const float*)d_in[6];  // [H]

    float* out   = (float*)d_out;
    float* hseq  = out;                                        // [B,T,H]
    float* hT    = out + (size_t)QB * QT * QH;                 // [B,H]
    float* cT    = hT + (size_t)QB * QH;                       // [B,H]
    float* xproj = (float*)d_ws;                               // [B,T,4] = 2 MB

    // Kernel 1: 131072 rows / (8 waves * 16 rows) = 1024 blocks
    qlstm_xproj_wmma<<<1024, 256, 0, stream>>>(x, W_in + QH * 4, xproj);

    // Kernel 2: one workgroup per batch element
    qlstm_recur<<<QB, 256, 0, stream>>>(xproj, W_in, b_in, Wq, bq, W_out, b_out,
                                        hseq, hT, cT);
}